// FeatureLearningNet_13331578487175
// MI455X (gfx1250) — compile-verified
//
#include <hip/hip_runtime.h>
#include <math.h>

// ---- problem constants (match reference) ----
#define T_PTS 35
#define C1_IN 7
#define C1_OUT 16
#define C2_IN 32
#define C2_OUT 64
#define GRID_D 10
#define GRID_H 400
#define GRID_W 352
#define EPSV 1e-5f

typedef __attribute__((ext_vector_type(2))) float v2f;
typedef __attribute__((ext_vector_type(4))) float f4;
typedef __attribute__((ext_vector_type(8))) float v8f;

// ---------------------------------------------------------------------------
// 0) zero the (poisoned) dense output grid with non-temporal b128 stores
// ---------------------------------------------------------------------------
__global__ void k_zero(float* __restrict__ out, long n) {
  const f4 z = {0.f, 0.f, 0.f, 0.f};
  long n4 = n >> 2;
  long stride = (long)gridDim.x * blockDim.x;
  for (long j = (long)blockIdx.x * blockDim.x + threadIdx.x; j < n4; j += stride)
    __builtin_nontemporal_store(z, ((f4*)out) + j);
  if (blockIdx.x == 0 && threadIdx.x == 0)
    for (long j = n4 << 2; j < n; ++j) out[j] = 0.f;
}

__global__ void k_zero_stats(float* __restrict__ ws) {
  if (threadIdx.x < 320) ws[threadIdx.x] = 0.f;
}

// ---------------------------------------------------------------------------
// 1) per-point validity mask: (max over 7 input channels) != 0
// ---------------------------------------------------------------------------
__global__ void k_mask(const float* __restrict__ feat, float* __restrict__ mask, int npts) {
  int p = blockIdx.x * blockDim.x + threadIdx.x;
  if (p >= npts) return;
  const float* f = feat + (size_t)p * C1_IN;
  float m = f[0];
#pragma unroll
  for (int c = 1; c < C1_IN; ++c) m = fmaxf(m, f[c]);
  mask[p] = (m != 0.f) ? 1.f : 0.f;
}

// ---------------------------------------------------------------------------
// 2) GEMM1: pw1 = relu(feature[M,7] @ W1[7,16] + b1), fp32 WMMA 16x16x4
//    A loads are fully unconditional (clamped addresses). OOB rows produce
//    garbage only in their own D rows, which the epilogue discards; the K=7
//    pad column multiplies B's zeroed pad row, contributing exactly 0.
// ---------------------------------------------------------------------------
__global__ void k_gemm1(const float* __restrict__ feat, const float* __restrict__ W1,
                        const float* __restrict__ b1, float* __restrict__ pw1,
                        float* __restrict__ gsum, float* __restrict__ gsumsq, int M) {
  __shared__ float sW[8 * 16];   // K padded 7->8, row-major [k][n], row 7 = zeros
  __shared__ float ssum[C1_OUT], ssq[C1_OUT];
  int tid = threadIdx.x;
  if (tid < C1_OUT) { ssum[tid] = 0.f; ssq[tid] = 0.f; }
  if (tid < 128) {
    int k = tid >> 4, n = tid & 15;
    sW[tid] = (k < C1_IN) ? W1[k * C1_OUT + n] : 0.f;
  }
  __syncthreads();

  int wave = tid >> 5, lane = tid & 31;
  int lg = lane >> 4;          // lane group: selects K pair
  int ln = lane & 15;
  int rowBase = (blockIdx.x * 4 + wave) * 16;
  int row = rowBase + ln;
  int rowc = row < M ? row : (M - 1);       // clamp: loads always in-bounds
  const float* fr = feat + (size_t)rowc * C1_IN;

  v8f acc = {0.f, 0.f, 0.f, 0.f, 0.f, 0.f, 0.f, 0.f};
#pragma unroll
  for (int s = 0; s < 2; ++s) {
    int k = s * 4 + lg * 2;                 // k in {0,2,4,6}: always < 7
    int k1 = (k + 1 < C1_IN) ? (k + 1) : (C1_IN - 1);  // clamp pad column
    v2f a, b;
    a.x = fr[k];                            // unconditional loads
    a.y = fr[k1];                           // pad col hits zero B row -> 0 contribution
    b.x = sW[k * 16 + ln];
    b.y = sW[(k + 1) * 16 + ln];
    acc = __builtin_amdgcn_wmma_f32_16x16x4_f32(false, a, false, b, (short)0, acc, false, false);
  }

  float bc = b1[ln];
  float ls = 0.f, lq = 0.f;
  if (rowBase + 16 <= M) {                  // full tile (8203/8204 blocks): no guards
#pragma unroll
    for (int r = 0; r < 8; ++r) {
      int orow = rowBase + r + lg * 8;      // C/D layout: M = r + 8*(lane>=16)
      float v = acc[r] + bc;
      v = v > 0.f ? v : 0.f;
      pw1[(size_t)orow * C1_OUT + ln] = v;
      ls += v; lq += v * v;
    }
  } else {                                  // edge tile: guard stores and stats
#pragma unroll
    for (int r = 0; r < 8; ++r) {
      int orow = rowBase + r + lg * 8;
      float v = acc[r] + bc;
      v = v > 0.f ? v : 0.f;
      if (orow < M) {
        pw1[(size_t)orow * C1_OUT + ln] = v;
        ls += v; lq += v * v;
      }
    }
  }
  atomicAdd(&ssum[ln], ls);
  atomicAdd(&ssq[ln], lq);
  __syncthreads();
  if (tid < C1_OUT) { atomicAdd(&gsum[tid], ssum[tid]); atomicAdd(&gsumsq[tid], ssq[tid]); }
}

// ---------------------------------------------------------------------------
// 3) finalize BN: scale = g*rsqrt(var+eps), shift = beta - mu*scale
// ---------------------------------------------------------------------------
__global__ void k_finalize(const float* __restrict__ gsum, const float* __restrict__ gsumsq,
                           const float* __restrict__ g, const float* __restrict__ be,
                           float* __restrict__ scale, float* __restrict__ shift,
                           int C, float invN) {
  int c = threadIdx.x;
  if (c >= C) return;
  float mu = gsum[c] * invN;
  float var = gsumsq[c] * invN - mu * mu;
  float sc = g[c] * rsqrtf(var + EPSV);
  scale[c] = sc;
  shift[c] = be[c] - mu * sc;
}

// ---------------------------------------------------------------------------
// 4) BN1 + max-over-T + broadcast-concat + mask: x1[k,t,0:16]=pwn*m,
//    x1[k,t,16:32]=maxT(pwn)*m.  One thread per (voxel, channel).
// ---------------------------------------------------------------------------
__global__ void k_bn1cat(const float* __restrict__ pw1, const float* __restrict__ mask,
                         const float* __restrict__ scale, const float* __restrict__ shift,
                         float* __restrict__ x1, int K) {
  int i = blockIdx.x * blockDim.x + threadIdx.x;
  if (i >= K * C1_OUT) return;
  int k = i >> 4, c = i & 15;
  float sc = scale[c], sh = shift[c];
  const float* pr = pw1 + (size_t)k * T_PTS * C1_OUT + c;
  float vals[T_PTS];
  float amax = -__builtin_inff();
#pragma unroll
  for (int t = 0; t < T_PTS; ++t) {
    float v = pr[(size_t)t * C1_OUT] * sc + sh;   // agg over ALL T (incl. padded), pre-mask
    vals[t] = v;
    amax = fmaxf(amax, v);
  }
  float* xw = x1 + (size_t)k * T_PTS * C2_IN + c;
  const float* mk = mask + (size_t)k * T_PTS;
#pragma unroll
  for (int t = 0; t < T_PTS; ++t) {
    float m = mk[t];
    xw[(size_t)t * C2_IN] = vals[t] * m;
    xw[(size_t)t * C2_IN + C1_OUT] = amax * m;
  }
}

// ---------------------------------------------------------------------------
// 5) GEMM2: pw2 = relu(x1[M,32] @ W2[32,64] + b2), fp32 WMMA 16x16x4
//    each wave: 16 rows x 64 cols = 4 accumulators, 8 K-steps => 32 wmma/wave
// ---------------------------------------------------------------------------
__global__ void k_gemm2(const float* __restrict__ x1, const float* __restrict__ W2,
                        const float* __restrict__ b2, float* __restrict__ pw2,
                        float* __restrict__ gsum, float* __restrict__ gsumsq, int M) {
  __shared__ float sW[C2_IN * C2_OUT];   // 2048 floats = 8 KB, row-major [k][n]
  __shared__ float ssum[C2_OUT], ssq[C2_OUT];
  int tid = threadIdx.x;
  for (int i = tid; i < C2_IN * C2_OUT; i += 128) sW[i] = W2[i];
  if (tid < C2_OUT) { ssum[tid] = 0.f; ssq[tid] = 0.f; }
  __syncthreads();

  int wave = tid >> 5, lane = tid & 31;
  int lg = lane >> 4, ln = lane & 15;
  int rowBase = (blockIdx.x * 4 + wave) * 16;
  int row = rowBase + ln;
  int rowc = row < M ? row : (M - 1);       // clamp: loads always in-bounds
  const float* xr = x1 + (size_t)rowc * C2_IN;

  v8f acc[4];
#pragma unroll
  for (int nt = 0; nt < 4; ++nt) acc[nt] = (v8f){0.f, 0.f, 0.f, 0.f, 0.f, 0.f, 0.f, 0.f};

#pragma unroll
  for (int s = 0; s < 8; ++s) {
    int k = s * 4 + lg * 2;
    v2f a = *(const v2f*)(xr + k);          // unconditional b64 load (8B aligned)
#pragma unroll
    for (int nt = 0; nt < 4; ++nt) {
      int n = nt * 16 + ln;
      v2f b;
      b.x = sW[k * C2_OUT + n];
      b.y = sW[(k + 1) * C2_OUT + n];
      acc[nt] = __builtin_amdgcn_wmma_f32_16x16x4_f32(false, a, false, b, (short)0, acc[nt], false, false);
    }
  }

  bool full = (rowBase + 16) <= M;          // wave-uniform
#pragma unroll
  for (int nt = 0; nt < 4; ++nt) {
    int col = nt * 16 + ln;
    float bc = b2[col];
    float ls = 0.f, lq = 0.f;
    if (full) {
#pragma unroll
      for (int r = 0; r < 8; ++r) {
        int orow = rowBase + r + lg * 8;
        float v = acc[nt][r] + bc;
        v = v > 0.f ? v : 0.f;
        pw2[(size_t)orow * C2_OUT + col] = v;
        ls += v; lq += v * v;
      }
    } else {
#pragma unroll
      for (int r = 0; r < 8; ++r) {
        int orow = rowBase + r + lg * 8;
        float v = acc[nt][r] + bc;
        v = v > 0.f ? v : 0.f;
        if (orow < M) {
          pw2[(size_t)orow * C2_OUT + col] = v;
          ls += v; lq += v * v;
        }
      }
    }
    atomicAdd(&ssum[col], ls);
    atomicAdd(&ssq[col], lq);
  }
  __syncthreads();
  if (tid < C2_OUT) { atomicAdd(&gsum[tid], ssum[tid]); atomicAdd(&gsumsq[tid], ssq[tid]); }
}

// ---------------------------------------------------------------------------
// 6) BN2 + max-over-T + concat-max + scatter-add into dense grid
//    voxelwise[k,c]    = max_t(pwn2*mask)
//    voxelwise[k,64+c] = max_t(maxT(pwn2)*mask)
// ---------------------------------------------------------------------------
__global__ void k_bn2scatter(const float* __restrict__ pw2, const float* __restrict__ mask,
                             const float* __restrict__ scale, const float* __restrict__ shift,
                             const int* __restrict__ coord, float* __restrict__ out, int K) {
  int i = blockIdx.x * blockDim.x + threadIdx.x;
  if (i >= K * C2_OUT) return;
  int k = i >> 6, c = i & 63;
  float sc = scale[c], sh = shift[c];
  const float* pr = pw2 + (size_t)k * T_PTS * C2_OUT + c;
  const float* mk = mask + (size_t)k * T_PTS;
  float amax = -__builtin_inff();
  float mmax = -__builtin_inff();
  bool anyInv = false;
#pragma unroll
  for (int t = 0; t < T_PTS; ++t) {
    float v = pr[(size_t)t * C2_OUT] * sc + sh;
    amax = fmaxf(amax, v);
    float m = mk[t];
    mmax = fmaxf(mmax, v * m);
    anyInv |= (m == 0.f);
  }
  float hi = anyInv ? fmaxf(amax, 0.f) : amax;   // max_t(agg*mask)
  int cb = coord[k * 4 + 0], cd = coord[k * 4 + 1];
  int ch = coord[k * 4 + 2], cw = coord[k * 4 + 3];
  size_t base = ((((size_t)cb * GRID_D + cd) * GRID_H + ch) * GRID_W + cw) * 128;
  atomicAdd(&out[base + c], mmax);       // duplicate coords sum (COO to_dense)
  atomicAdd(&out[base + C2_OUT + c], hi);
}

// ---------------------------------------------------------------------------
extern "C" void kernel_launch(void* const* d_in, const int* in_sizes, int n_in,
                              void* d_out, int out_size, void* d_ws, size_t ws_size,
                              hipStream_t stream) {
  const float* feat = (const float*)d_in[0];
  const int*   coord = (const int*)d_in[1];
  // d_in[2] = bs (always 1; batch index comes from coordinate anyway)
  const float* W1 = (const float*)d_in[3];
  const float* b1 = (const float*)d_in[4];
  const float* g1 = (const float*)d_in[5];
  const float* be1 = (const float*)d_in[6];
  const float* W2 = (const float*)d_in[7];
  const float* b2 = (const float*)d_in[8];
  const float* g2 = (const float*)d_in[9];
  const float* be2 = (const float*)d_in[10];

  int K = in_sizes[0] / (T_PTS * C1_IN);   // 15000
  int M = K * T_PTS;                       // 525000 points

  // workspace layout (floats)
  float* ws = (float*)d_ws;
  float* gsum1  = ws;        float* gsumsq1 = ws + 16;
  float* scale1 = ws + 32;   float* shift1  = ws + 48;
  float* gsum2  = ws + 64;   float* gsumsq2 = ws + 128;
  float* scale2 = ws + 192;  float* shift2  = ws + 256;  // stats end at 320
  float* mask = ws + 320;                       // M floats (offset even -> x1 8B aligned)
  float* x1   = mask + (size_t)M;               // M*32 floats
  float* pw1  = x1 + (size_t)M * C2_IN;         // M*16 floats
  float* pw2  = pw1;                            // reuse: pw1 dead before pw2 written (M*64)

  k_zero<<<2048, 256, 0, stream>>>((float*)d_out, (long)out_size);
  k_zero_stats<<<1, 320, 0, stream>>>(ws);
  k_mask<<<(M + 255) / 256, 256, 0, stream>>>(feat, mask, M);

  int gblocks = (M + 63) / 64;   // 4 waves/block, 16 rows/wave
  k_gemm1<<<gblocks, 128, 0, stream>>>(feat, W1, b1, pw1, gsum1, gsumsq1, M);
  k_finalize<<<1, 64, 0, stream>>>(gsum1, gsumsq1, g1, be1, scale1, shift1, C1_OUT, 1.f / (float)M);
  k_bn1cat<<<(K * C1_OUT + 255) / 256, 256, 0, stream>>>(pw1, mask, scale1, shift1, x1, K);

  k_gemm2<<<gblocks, 128, 0, stream>>>(x1, W2, b2, pw2, gsum2, gsumsq2, M);
  k_finalize<<<1, 64, 0, stream>>>(gsum2, gsumsq2, g2, be2, scale2, shift2, C2_OUT, 1.f / (float)M);
  k_bn2scatter<<<(K * C2_OUT + 255) / 256, 256, 0, stream>>>(pw2, mask, scale2, shift2, coord,
                                                             (float*)d_out, K);
}